// H_4269197492356
// MI455X (gfx1250) — compile-verified
//
#include <hip/hip_runtime.h>

typedef float v2f __attribute__((ext_vector_type(2)));
typedef float v8f __attribute__((ext_vector_type(8)));

#define LDST 130   // 128 + 2 pad: row stride bumps bank by 2 -> conflict-free column reads

__global__ __launch_bounds__(256)
void fused_block_bilinear(const float* __restrict__ x,
                          const int* __restrict__ perm,
                          const float* __restrict__ hr1, const float* __restrict__ hi1,
                          const float* __restrict__ hr2, const float* __restrict__ hi2,
                          const float* __restrict__ hr3, const float* __restrict__ hi3,
                          const float* __restrict__ hr4, const float* __restrict__ hi4,
                          float* __restrict__ out)
{
    __shared__ float Xs[128 * LDST];   // current X_re or X_im block (p,a)
    __shared__ float Ts[128 * LDST];   // per-term intermediate T = X * R
    __shared__ int   sp[16];

    const int tid = threadIdx.x;
    const int wg  = blockIdx.x;        // 0..255
    const int ab  = wg & 7;            // column block a
    const int pb  = (wg >> 3) & 7;     // row block p
    const int bb  = wg >> 6;           // batch

    if (tid < 16) sp[tid] = perm[tid];
    __syncthreads();

    const int w  = tid >> 5;           // wave id 0..7
    const int l  = tid & 31;
    const int lr = l & 15;             // lane-row / lane-col within fragment
    const int kh = (l >> 4) << 1;      // K offset: 0 (lanes 0-15) or 2 (lanes 16-31)

    const size_t blkA = (size_t)ab * (128 * 128);
    const size_t blkP = (size_t)pb * (128 * 128);

    v8f zero;
#pragma unroll
    for (int i = 0; i < 8; ++i) zero[i] = 0.0f;

    v8f accRe[8], accIm[8];
#pragma unroll
    for (int t = 0; t < 8; ++t) { accRe[t] = zero; accIm[t] = zero; }

    // Gather one 128x128 block (re: phase=0, im: phase=1) through the row/col
    // permutation into LDS. Consecutive lanes -> consecutive (stride-2) floats.
    auto loadX = [&](int phase) {
        const size_t bbase = (size_t)bb * (1024u * 1024u * 2u);
        for (int it = 0; it < 64; ++it) {
            const int e  = it * 256 + tid;
            const int j  = e >> 7;
            const int j2 = e & 127;
            const int grow = sp[2 * pb + (j  >> 6)] * 64 + (j  & 63);
            const int gcol = sp[2 * ab + (j2 >> 6)] * 64 + (j2 & 63);
            Xs[j * LDST + j2] = x[bbase + ((size_t)grow * 1024 + (size_t)gcol) * 2 + phase];
        }
    };

    // One bilinear term:  acc += L^T * (Xs * R)
    auto term = [&](const float* __restrict__ R, const float* __restrict__ L, v8f* acc) {
        __syncthreads();                       // Ts free / Xs ready
        // -------- stage 1: Ts = Xs * R --------
        // Wave w owns output COLUMNS [16w, 16w+16) across all 8 row strips, so
        // the B fragment (R) is fixed per k-step: each R element is loaded from
        // global exactly once per workgroup. A fragments stream from LDS.
        {
            v8f c[8];
#pragma unroll
            for (int it = 0; it < 8; ++it) c[it] = zero;
            const float* Xcol = &Xs[lr * LDST];           // + 16*it*LDST + k
            for (int k0 = 0; k0 < 128; k0 += 4) {
                const int k = k0 + kh;
                v2f bv;                                    // B[K][N=16w+lr]
                bv.x = R[(size_t)k * 128 + 16 * w + lr];
                bv.y = R[(size_t)(k + 1) * 128 + 16 * w + lr];
#pragma unroll
                for (int it = 0; it < 8; ++it) {
                    v2f av;                                // A[M=lr][K=k..k+1] of row strip it
                    av.x = Xcol[(16 * it) * LDST + k];
                    av.y = Xcol[(16 * it) * LDST + k + 1];
                    c[it] = __builtin_amdgcn_wmma_f32_16x16x4_f32(
                        false, av, false, bv, (short)0, c[it], false, false);
                }
            }
#pragma unroll
            for (int it = 0; it < 8; ++it)
#pragma unroll
                for (int v = 0; v < 8; ++v)     // D: vgpr v = row M=v (+8 for lanes 16-31)
                    Ts[(16 * it + v + 4 * kh) * LDST + 16 * w + lr] = c[it][v];
        }
        __syncthreads();
        // -------- stage 2: acc += L^T * Ts (wave w -> out rows [16w, 16w+16)) --------
        for (int k0 = 0; k0 < 128; k0 += 4) {
            const int k = k0 + kh;
            v2f av;                             // A[M][K] = L[K][16w+M]
            av.x = L[(size_t)k * 128 + 16 * w + lr];
            av.y = L[(size_t)(k + 1) * 128 + 16 * w + lr];
            const float* t0 = &Ts[k * LDST + lr];
            const float* t1 = t0 + LDST;
#pragma unroll
            for (int jt = 0; jt < 8; ++jt) {
                v2f bv;
                bv.x = t0[16 * jt];
                bv.y = t1[16 * jt];
                acc[jt] = __builtin_amdgcn_wmma_f32_16x16x4_f32(
                    false, av, false, bv, (short)0, acc[jt], false, false);
            }
        }
    };

    // Phase 1: X_re resident
    loadX(0);
    term(hr1 + blkA, hr1 + blkP, accRe);   // hr1^T Xre hr1
    term(hi1 + blkA, hi1 + blkP, accRe);   // hi1^T Xre hi1
    term(hi4 + blkA, hr4 + blkP, accIm);   // hr4^T Xre hi4
    term(hr4 + blkA, hi4 + blkP, accIm);   // hi4^T Xre hr4
    // Phase 2: X_im resident
    loadX(1);
    term(hi2 + blkA, hr2 + blkP, accRe);   // hr2^T Xim hi2
    term(hr2 + blkA, hi2 + blkP, accRe);   // hi2^T Xim hr2
    term(hr3 + blkA, hr3 + blkP, accIm);   // hr3^T Xim hr3
    term(hi3 + blkA, hi3 + blkP, accIm);   // hi3^T Xim hi3

    // Store interleaved (re,im) as 8-byte packed writes; output is NOT un-permuted.
    float2* out2 = reinterpret_cast<float2*>(out);
#pragma unroll
    for (int jt = 0; jt < 8; ++jt) {
#pragma unroll
        for (int v = 0; v < 8; ++v) {
            const int row = pb * 128 + 16 * w + v + 4 * kh;
            const int col = ab * 128 + 16 * jt + lr;
            float2 o;
            o.x = accRe[jt][v];
            o.y = accIm[jt][v];
            out2[((size_t)bb * 1024 + row) * 1024 + col] = o;
        }
    }
}

extern "C" void kernel_launch(void* const* d_in, const int* in_sizes, int n_in,
                              void* d_out, int out_size, void* d_ws, size_t ws_size,
                              hipStream_t stream) {
    (void)in_sizes; (void)n_in; (void)out_size; (void)d_ws; (void)ws_size;
    const float* x    = (const float*)d_in[0];
    const int*   perm = (const int*)  d_in[1];
    const float* hr1  = (const float*)d_in[2];
    const float* hi1  = (const float*)d_in[3];
    const float* hr2  = (const float*)d_in[4];
    const float* hi2  = (const float*)d_in[5];
    const float* hr3  = (const float*)d_in[6];
    const float* hi3  = (const float*)d_in[7];
    const float* hr4  = (const float*)d_in[8];
    const float* hi4  = (const float*)d_in[9];
    float* out = (float*)d_out;

    dim3 grid(256), block(256);   // one WG per (b, p, a) 128x128 output block
    hipLaunchKernelGGL(fused_block_bilinear, grid, block, 0, stream,
                       x, perm, hr1, hi1, hr2, hi2, hr3, hi3, hr4, hi4, out);
}